// SE3MovingFrame_42511586296167
// MI455X (gfx1250) — compile-verified
//
#include <hip/hip_runtime.h>
#include <math.h>

typedef __attribute__((ext_vector_type(2))) float v2f;
typedef __attribute__((ext_vector_type(8))) float v8f;

#define SIGMA   2.0f
#define KW      8          // half width
#define KLEN    17         // kernel taps
#define NPOS    64         // length of each conv axis
#define NLINES  16384      // 64*64*4 lines per pass

// Compute the 3 Gaussian-derivative kernels into LDS (g[3][17]).
__device__ __forceinline__ void compute_g(float* g) {
    int t = (int)threadIdx.x;
    if (t < KLEN) {
        float x  = (float)(t - KW);
        float g0 = expf(-x * x / (2.f * SIGMA * SIGMA)) / sqrtf(6.283185307179586f * SIGMA);
        float tt = x / SIGMA;
        g[0 * KLEN + t] = g0;
        g[1 * KLEN + t] = -tt * g0 / SIGMA;
        g[2 * KLEN + t] = (tt * tt - 1.f) * g0 / (SIGMA * SIGMA);
    }
}

// WMMA separable-conv pass along one axis.
//   line address of line l = (l / LB) * HS + (l % LB); element = line_addr + pos * S
//   out[k] = correlate(in, g_k) along the axis, zero ('SAME') padding.
// One wave handles a 16-line x 16-position tile. D = A(kernel band, 16x32) * B(data, 32x16),
// accumulated as 8 chained V_WMMA_F32_16X16X4_F32.
__global__ __launch_bounds__(256) void conv_wmma(
    const float* __restrict__ in,
    float* __restrict__ out0, float* __restrict__ out1, float* __restrict__ out2,
    int nk, int S, int LB, int HS)
{
    __shared__ float g[3 * KLEN];
    compute_g(g);
    __syncthreads();

    const int lane = (int)threadIdx.x & 31;
    const int half = lane >> 4;     // 0 or 1
    const int lm   = lane & 15;

    const int uid      = (int)blockIdx.x * 8 + ((int)threadIdx.x >> 5);
    const int posTile  = uid & 3;          // 64/16 = 4 position tiles
    const int lineTile = uid >> 2;         // 16384/16 = 1024 line tiles
    const int l0 = lineTile * 16;
    const int x0 = posTile * 16;

    // B-matrix (data) column n = lm -> data line l0+lm (contiguous across lanes 0-15)
    const int ldata    = l0 + lm;
    const int lineAddr = (ldata / LB) * HS + (ldata % LB);

    v8f acc0 = {0.f, 0.f, 0.f, 0.f, 0.f, 0.f, 0.f, 0.f};
    v8f acc1 = acc0;
    v8f acc2 = acc0;

#pragma unroll
    for (int c = 0; c < 8; ++c) {
        // B chunk (4x16 data): row j = 4c + r + 2*half, col = lm
        v2f b;
#pragma unroll
        for (int r = 0; r < 2; ++r) {
            int j = 4 * c + r + 2 * half;
            int x = x0 + j - KW;
            float v = 0.f;
            if (x >= 0 && x < NPOS) v = in[lineAddr + x * S];
            b[r] = v;
        }
        // A chunks (16x4 banded kernel): A[m=lm, j] = g_k[j - lm]
        {
            v2f a;
#pragma unroll
            for (int r = 0; r < 2; ++r) {
                int idx = 4 * c + r + 2 * half - lm;
                a[r] = (idx >= 0 && idx < KLEN) ? g[0 * KLEN + idx] : 0.f;
            }
            acc0 = __builtin_amdgcn_wmma_f32_16x16x4_f32(false, a, false, b, (short)0, acc0, false, false);
        }
        if (nk > 1) {
            v2f a;
#pragma unroll
            for (int r = 0; r < 2; ++r) {
                int idx = 4 * c + r + 2 * half - lm;
                a[r] = (idx >= 0 && idx < KLEN) ? g[1 * KLEN + idx] : 0.f;
            }
            acc1 = __builtin_amdgcn_wmma_f32_16x16x4_f32(false, a, false, b, (short)0, acc1, false, false);
        }
        if (nk > 2) {
            v2f a;
#pragma unroll
            for (int r = 0; r < 2; ++r) {
                int idx = 4 * c + r + 2 * half - lm;
                a[r] = (idx >= 0 && idx < KLEN) ? g[2 * KLEN + idx] : 0.f;
            }
            acc2 = __builtin_amdgcn_wmma_f32_16x16x4_f32(false, a, false, b, (short)0, acc2, false, false);
        }
    }

    // D layout: vgpr r, lane: M (output position) = r + 8*half, N (line) = lm
#pragma unroll
    for (int r = 0; r < 8; ++r) {
        int xo = x0 + r + 8 * half;
        int addr = lineAddr + xo * S;
        out0[addr] = acc0[r];
        if (nk > 1) out1[addr] = acc1[r];
        if (nk > 2) out2[addr] = acc2[r];
    }
}

// Eigenvector of symmetric 3x3 (H - l*I) via largest cross-product of rows.
__device__ __forceinline__ void eigvec3(
    float h00, float h01, float h02, float h11, float h12, float h22,
    float l, float v[3])
{
    float r0x = h00 - l, r0y = h01,     r0z = h02;
    float r1x = h01,     r1y = h11 - l, r1z = h12;
    float r2x = h02,     r2y = h12,     r2z = h22 - l;

    float ax = r0y * r1z - r0z * r1y, ay = r0z * r1x - r0x * r1z, az = r0x * r1y - r0y * r1x;
    float bx = r0y * r2z - r0z * r2y, by = r0z * r2x - r0x * r2z, bz = r0x * r2y - r0y * r2x;
    float cx = r1y * r2z - r1z * r2y, cy = r1z * r2x - r1x * r2z, cz = r1x * r2y - r1y * r2x;

    float na = ax * ax + ay * ay + az * az;
    float nb = bx * bx + by * by + bz * bz;
    float nc = cx * cx + cy * cy + cz * cz;

    float vx = ax, vy = ay, vz = az, n = na;
    if (nb > n) { vx = bx; vy = by; vz = bz; n = nb; }
    if (nc > n) { vx = cx; vy = cy; vz = cz; n = nc; }
    if (n < 1e-30f) { vx = 1.f; vy = 0.f; vz = 0.f; n = 1.f; }
    float inv = 1.f / sqrtf(n);
    v[0] = vx * inv; v[1] = vy * inv; v[2] = vz * inv;
}

// Fused z-axis conv (9 derivative fields from 6 intermediates) + per-voxel eigen frame.
// One block per (x,y): 256 threads = 64 z * 4 channels.
__global__ __launch_bounds__(256) void frame_kernel(
    const float* __restrict__ s00, const float* __restrict__ s01, const float* __restrict__ s02,
    const float* __restrict__ s10, const float* __restrict__ s11, const float* __restrict__ s20,
    float* __restrict__ out)
{
    __shared__ float g[3 * KLEN];
    __shared__ float tile[6][80 * 4];   // z halo [-8, 71] x 4 channels

    compute_g(g);

    const int xy   = (int)blockIdx.x;      // 0..4095 = x*64 + y
    const int base = xy * 256;             // element offset of (x, y, z=0, c=0)

    const float* F[6] = {s00, s01, s02, s10, s11, s20};
    for (int i = (int)threadIdx.x; i < 6 * 320; i += 256) {
        int f  = i / 320;
        int zz = (i % 320) >> 2;
        int c  = i & 3;
        int z  = zz - 8;
        float v = 0.f;
        if (z >= 0 && z < 64) v = F[f][base + z * 4 + c];
        tile[f][zz * 4 + c] = v;
    }
    __syncthreads();

    const int c = (int)threadIdx.x & 3;
    const int z = (int)threadIdx.x >> 2;

    float ux = 0, uy = 0, uz = 0, uxx = 0, uyy = 0, uzz = 0, uxy = 0, uxz = 0, uyz = 0;
#pragma unroll
    for (int t = 0; t < KLEN; ++t) {
        int idx = (z + t) * 4 + c;          // zz = z + t  (z-8+t, rebased by +8)
        float a00 = tile[0][idx];
        float a01 = tile[1][idx];
        float a02 = tile[2][idx];
        float a10 = tile[3][idx];
        float a11 = tile[4][idx];
        float a20 = tile[5][idx];
        float k0 = g[t], k1 = g[KLEN + t], k2 = g[2 * KLEN + t];
        uz  += k1 * a00;  uzz += k2 * a00;
        uy  += k0 * a01;  uyz += k1 * a01;
        uyy += k0 * a02;
        ux  += k0 * a10;  uxz += k1 * a10;
        uxy += k0 * a11;
        uxx += k0 * a20;
    }

    // ----- 3x3 symmetric eigendecomposition (descending) -----
    float h00 = uxx, h01 = uxy, h02 = uxz, h11 = uyy, h12 = uyz, h22 = uzz;
    float q  = (h00 + h11 + h22) * (1.f / 3.f);
    float p1 = h01 * h01 + h02 * h02 + h12 * h12;
    float b00 = h00 - q, b11 = h11 - q, b22 = h22 - q;
    float p2 = b00 * b00 + b11 * b11 + b22 * b22 + 2.f * p1;
    float p  = sqrtf(p2 * (1.f / 6.f));
    float invp = (p > 1e-20f) ? 1.f / p : 0.f;
    float c00 = b00 * invp, c11 = b11 * invp, c22 = b22 * invp;
    float c01 = h01 * invp, c02 = h02 * invp, c12 = h12 * invp;
    float detB = c00 * (c11 * c22 - c12 * c12)
               - c01 * (c01 * c22 - c12 * c02)
               + c02 * (c01 * c12 - c11 * c02);
    float r = fminf(1.f, fmaxf(-1.f, 0.5f * detB));
    float phi = acosf(r) * (1.f / 3.f);
    float l0 = q + 2.f * p * cosf(phi);
    float l2 = q + 2.f * p * cosf(phi + 2.0943951023931953f);   // + 2*pi/3
    float l1 = 3.f * q - l0 - l2;

    float v0[3], v1[3], v2[3];
    eigvec3(h00, h01, h02, h11, h12, h22, l0, v0);
    eigvec3(h00, h01, h02, h11, h12, h22, l1, v1);

    // Orthogonalize v1 against v0 (robust in near-degenerate cases)
    float d01 = v1[0] * v0[0] + v1[1] * v0[1] + v1[2] * v0[2];
    v1[0] -= d01 * v0[0]; v1[1] -= d01 * v0[1]; v1[2] -= d01 * v0[2];
    float n1 = v1[0] * v1[0] + v1[1] * v1[1] + v1[2] * v1[2];
    if (n1 < 1e-20f) {
        float tx = 1.f, ty = 0.f, tz = 0.f;
        if (fabsf(v0[0]) > 0.9f) { tx = 0.f; ty = 1.f; }
        v1[0] = v0[1] * tz - v0[2] * ty;
        v1[1] = v0[2] * tx - v0[0] * tz;
        v1[2] = v0[0] * ty - v0[1] * tx;
        n1 = v1[0] * v1[0] + v1[1] * v1[1] + v1[2] * v1[2];
        if (n1 < 1e-30f) { v1[0] = 0.f; v1[1] = 1.f; v1[2] = 0.f; n1 = 1.f; }
    }
    float in1 = 1.f / sqrtf(n1);
    v1[0] *= in1; v1[1] *= in1; v1[2] *= in1;

    // Sign fixes against gradient
    float dot0 = v0[0] * ux + v0[1] * uy + v0[2] * uz;
    if (dot0 < 0.f) { v0[0] = -v0[0]; v0[1] = -v0[1]; v0[2] = -v0[2]; }
    float dot1 = v1[0] * ux + v1[1] * uy + v1[2] * uz;
    if (dot1 < 0.f) { v1[0] = -v1[0]; v1[1] = -v1[1]; v1[2] = -v1[2]; }

    // det>0 fix for orthonormal frame == v2 = v0 x v1
    v2[0] = v0[1] * v1[2] - v0[2] * v1[1];
    v2[1] = v0[2] * v1[0] - v0[0] * v1[2];
    v2[2] = v0[0] * v1[1] - v0[1] * v1[0];

    float mask = fabsf(l0 - l1);
    mask = fminf(mask, fabsf(l1 - l2));
    mask = fminf(mask, fabsf(dot0));
    mask = fminf(mask, fabsf(dot1));
    if (mask > 1e-5f) mask = 1.f;

    int o = (base + z * 4 + c) * 9;
    out[o + 0] = v0[0] * mask; out[o + 1] = v1[0] * mask; out[o + 2] = v2[0] * mask;
    out[o + 3] = v0[1] * mask; out[o + 4] = v1[1] * mask; out[o + 5] = v2[1] * mask;
    out[o + 6] = v0[2] * mask; out[o + 7] = v1[2] * mask; out[o + 8] = v2[2] * mask;
}

extern "C" void kernel_launch(void* const* d_in, const int* in_sizes, int n_in,
                              void* d_out, int out_size, void* d_ws, size_t ws_size,
                              hipStream_t stream) {
    const float* in = (const float*)d_in[0];
    float* out = (float*)d_out;
    float* ws  = (float*)d_ws;

    const size_t NV = 64ull * 64 * 64 * 4;   // 1,048,576 floats per field
    float* A0  = ws + 0 * NV;   // conv_x(u, g0)
    float* A1  = ws + 1 * NV;   // conv_x(u, g1)
    float* A2  = ws + 2 * NV;   // conv_x(u, g2)
    float* B00 = ws + 3 * NV;   // conv_y(A0, g0)
    float* B01 = ws + 4 * NV;   // conv_y(A0, g1)
    float* B02 = ws + 5 * NV;   // conv_y(A0, g2)
    float* B10 = ws + 6 * NV;   // conv_y(A1, g0)
    float* B11 = ws + 7 * NV;   // conv_y(A1, g1)
    float* B20 = ws + 8 * NV;   // conv_y(A2, g0)

    // 4096 wave-units per field / 8 waves per block = 512 blocks
    // Pass X: axis stride 16384, all 16384 lines contiguous (LB=16384, HS=0)
    conv_wmma<<<512, 256, 0, stream>>>(in, A0, A1, A2, 3, 16384, 16384, 0);
    // Pass Y: axis stride 256, lines: addr = (l/256)*16384 + l%256
    conv_wmma<<<512, 256, 0, stream>>>(A0, B00, B01, B02, 3, 256, 256, 16384);
    conv_wmma<<<512, 256, 0, stream>>>(A1, B10, B11, B11, 2, 256, 256, 16384);
    conv_wmma<<<512, 256, 0, stream>>>(A2, B20, B20, B20, 1, 256, 256, 16384);
    // Pass Z fused with eigen frame computation
    frame_kernel<<<4096, 256, 0, stream>>>(B00, B01, B02, B10, B11, B20, out);
}